// WGP_84018150245011
// MI455X (gfx1250) — compile-verified
//
#include <hip/hip_runtime.h>
#include <math.h>

// MI455X / gfx1250 (wave32, WMMA, async global->LDS, 320KB LDS/WGP).
// out[b,c] = tanh( sum_{k<49} x[b,c,k]*W[c,k] + bias[c] ), B=4096, C=512.
// Roofline: 411 MB of x @ 23.3 TB/s => ~17.7 us floor; 0.21 GFLOP is free.
// Data path: contiguous async DMA of each 16x16 (batch x channel) tile into
// LDS (bypasses VGPRs, ASYNCcnt), bank-conflict-free transposed ds reads into
// V_WMMA_F32_16X16X4_F32 A-fragments, weight vector broadcast across all 16
// N-columns (the contraction is a per-channel GEMV; the 16x column redundancy
// is free under the memory roofline).

typedef float v2f __attribute__((ext_vector_type(2), aligned(4)));  // 4B-safe
typedef float v4f __attribute__((ext_vector_type(4)));
typedef float v8f __attribute__((ext_vector_type(8)));
typedef int   v4i __attribute__((vector_size(16)));                 // builtin arg type

typedef __attribute__((address_space(1))) v4i* g_v4i_p;   // global
typedef __attribute__((address_space(3))) v4i* l_v4i_p;   // LDS

#define CSIZE 512
#define KSZ   49              // 7*7
#define BATCH 4096
#define ROWF  784             // 16 channels * 49 floats, contiguous per batch
#define LDSTRIDE 820          // pad: 820 % 64 == 52 (bank spread), 820 % 4 == 0 (16B align)

#if __has_builtin(__builtin_amdgcn_global_load_async_to_lds_b128)
#define HAVE_ASYNC_LDS 1
#else
#define HAVE_ASYNC_LDS 0
#endif

__global__ __launch_bounds__(256) void conv1x1ch_tanh_wmma(
    const float* __restrict__ x,     // [BATCH, CSIZE, 49]
    const float* __restrict__ W,     // [CSIZE, 49]
    const float* __restrict__ bias,  // [CSIZE]
    float* __restrict__ out)         // [BATCH, CSIZE]
{
    __shared__ float xs[16 * LDSTRIDE];     // 52,480 B

    const int tid  = threadIdx.x;
    const int lane = tid & 31;
    const int wave = tid >> 5;

    const int tile = blockIdx.x;            // 0 .. 8191
    const int b0   = (tile >> 5) << 4;      // batch-tile base
    const int c0   = (tile & 31) << 4;      // channel-tile base

    // ---- Stage: 16 batch-rows of 784 contiguous floats -> LDS (row stride 820)
    // 3136 b128 chunks total; thread t copies chunks t, t+256, ...
    for (int q = tid; q < 16 * (ROWF / 4); q += 256) {
        const int r  = q / (ROWF / 4);          // batch row 0..15
        const int j4 = q - r * (ROWF / 4);      // b128 chunk within row
        const float* src = x + ((size_t)(b0 + r) * CSIZE + c0) * KSZ + 4 * j4;
        float*       dst = &xs[r * LDSTRIDE + 4 * j4];
#if HAVE_ASYNC_LDS
        __builtin_amdgcn_global_load_async_to_lds_b128(
            (g_v4i_p)src, (l_v4i_p)dst, /*offset=*/0, /*cpol=*/0);
#else
        *(v4f*)dst = *(const v4f*)src;          // global_load_b128 + ds_store_b128
#endif
    }
#if HAVE_ASYNC_LDS
#if __has_builtin(__builtin_amdgcn_s_wait_asynccnt)
    __builtin_amdgcn_s_wait_asynccnt(0);
#else
    asm volatile("s_wait_asynccnt 0" ::: "memory");
#endif
#endif
    __syncthreads();

    // ---- Compute: each wave owns 2 of the tile's 16 channels.
    const int m  = lane & 15;   // A-matrix row (batch within tile)
    const int p  = lane >> 4;   // half-wave selects K sub-pair {0,1} vs {2,3}
    const int kb = p << 1;

#pragma unroll
    for (int t = 0; t < 2; ++t) {
        const int cl = (wave << 1) + t;         // 0..15 within tile
        const int c  = c0 + cl;
        const float* wc  = W + (size_t)c * KSZ; // broadcast, L0-resident
        const int    lx0 = m * LDSTRIDE + cl * KSZ + kb;

        v8f acc = {0.f, 0.f, 0.f, 0.f, 0.f, 0.f, 0.f, 0.f};

        // k = 0..47 : 12 chained V_WMMA_F32_16X16X4_F32 fed from LDS
#pragma unroll
        for (int j = 0; j < 12; ++j) {
            v2f a  = *(const v2f*)(&xs[lx0 + 4 * j]);   // conflict-free ds reads
            v2f bw = *(const v2f*)(wc + 4 * j + kb);
            acc = __builtin_amdgcn_wmma_f32_16x16x4_f32(
                /*neg_a=*/false, a, /*neg_b=*/false, bw,
                /*c_mod=*/(short)0, acc, /*reuse_a=*/false, /*reuse_b=*/false);
        }

        // Tail chunk: only k=48 valid (p==0 half); zero-pad the rest.
        {
            float xv = xs[m * LDSTRIDE + cl * KSZ + 48];
            float wv = wc[48];
            v2f a  = { p ? 0.f : xv, 0.f };
            v2f bw = { p ? 0.f : wv, 0.f };
            acc = __builtin_amdgcn_wmma_f32_16x16x4_f32(
                false, a, false, bw, (short)0, acc, false, false);
        }

        // D: VGPR r holds row (r + 8*p); all 16 columns identical (w broadcast)
        // -> lanes 0 and 16 store the 16 outputs of this channel.
        if (m == 0) {
            const float bb = bias[c];
            float* op = out + (size_t)(b0 + (p << 3)) * CSIZE + c;
#pragma unroll
            for (int r = 0; r < 8; ++r) {
                op[(size_t)r * CSIZE] = tanhf(acc[r] + bb);
            }
        }
    }
}

extern "C" void kernel_launch(void* const* d_in, const int* in_sizes, int n_in,
                              void* d_out, int out_size, void* d_ws, size_t ws_size,
                              hipStream_t stream) {
    const float* x    = (const float*)d_in[0];  // (4096, 512, 7, 7) f32
    const float* W    = (const float*)d_in[1];  // (512, 7, 7) f32
    const float* bias = (const float*)d_in[2];  // (512,) f32
    float* out        = (float*)d_out;          // (4096, 512) f32

    const int tiles = (BATCH / 16) * (CSIZE / 16);   // 8192 blocks
    conv1x1ch_tanh_wmma<<<tiles, 256, 0, stream>>>(x, W, bias, out);
}